// DGCNN_87067577025190
// MI455X (gfx1250) — compile-verified
//
#include <hip/hip_runtime.h>
#include <hip/hip_bf16.h>

#define BB 8
#define PP 1024
#define KK 20
#define EPSF 1e-5f

typedef __attribute__((ext_vector_type(2))) float v2f;
typedef __attribute__((ext_vector_type(8))) float v8f;

// ---------------------------------------------------------------------------
// Pad pos [8192,3] -> XP [8192,4] (4th column zero) so all GEMM K % 4 == 0.
// ---------------------------------------------------------------------------
__global__ void dg_pad_pos_kernel(const float* __restrict__ pos, float* __restrict__ XP) {
    int i = blockIdx.x * blockDim.x + threadIdx.x;
    if (i >= BB * PP) return;
    XP[(size_t)i * 4 + 0] = pos[(size_t)i * 3 + 0];
    XP[(size_t)i * 4 + 1] = pos[(size_t)i * 3 + 1];
    XP[(size_t)i * 4 + 2] = pos[(size_t)i * 3 + 2];
    XP[(size_t)i * 4 + 3] = 0.f;
}

// ---------------------------------------------------------------------------
// Row squared norms: sq[i] = sum_c X[i][c]^2  (padded columns are zero)
// ---------------------------------------------------------------------------
__global__ void dg_sqnorm_kernel(const float* __restrict__ X, int lda, int Cin,
                                 float* __restrict__ sq) {
    int i = blockIdx.x * blockDim.x + threadIdx.x;
    if (i >= BB * PP) return;
    const float* r = X + (size_t)i * lda;
    float s = 0.f;
    for (int c = 0; c < Cin; ++c) s += r[c] * r[c];
    sq[i] = s;
}

// ---------------------------------------------------------------------------
// Generic B-operand pack into WMMA f32 16x16x4 register layout.
//   Bp[((nt*S + s)*32 + lane)*2 + j] = src[k][n] (- sub[k][n]), k = 4s+j+2*(lane>>4),
//   n = 16*nt + (lane&15); zero when k >= Kvalid.
// ---------------------------------------------------------------------------
__global__ void dg_bpack_kernel(const float* __restrict__ Bsrc,
                                const float* __restrict__ Bsub, int ldb,
                                int Kvalid, int S, int N,
                                float* __restrict__ Bp) {
    int t = blockIdx.x * blockDim.x + threadIdx.x;
    int total = (N >> 4) * S * 64;
    if (t >= total) return;
    int j = t & 1;
    int l = (t >> 1) & 31;
    int rem = t >> 6;
    int s  = rem % S;
    int nt = rem / S;
    int k = 4 * s + j + 2 * (l >> 4);
    int n = nt * 16 + (l & 15);
    float v = 0.f;
    if (k < Kvalid) {
        v = Bsrc[(size_t)k * ldb + n];
        if (Bsub) v -= Bsub[(size_t)k * ldb + n];
    }
    Bp[t] = v;
}

// ---------------------------------------------------------------------------
// Pack X (per cloud) into gram B-operand layout: B[k][q] = X[q][k].
//   GB[(((b*64 + qt)*S + s)*32 + lane)*2 + j]
// ---------------------------------------------------------------------------
__global__ void dg_grampack_kernel(const float* __restrict__ X, int lda, int S,
                                   float* __restrict__ GB) {
    int t = blockIdx.x * blockDim.x + threadIdx.x;
    int total = BB * 64 * S * 64;
    if (t >= total) return;
    int j = t & 1;
    int l = (t >> 1) & 31;
    int rem = t >> 6;
    int s  = rem % S;  rem /= S;
    int qt = rem % 64;
    int b  = rem / 64;
    int k = 4 * s + j + 2 * (l >> 4);
    int q = qt * 16 + (l & 15);
    GB[t] = X[((size_t)b * PP + q) * lda + k];
}

// ---------------------------------------------------------------------------
// Pairwise distances via WMMA Gram tiles. Each wave computes a 16x64 strip
// (4 q-tiles) so one A load feeds 4 WMMAs per K-step.
//   D[b][p][q] = sq[p] + sq[q] - 2 * dot(X[p], X[q])
// ---------------------------------------------------------------------------
__global__ void dg_gram_dist_kernel(const float* __restrict__ X, int lda, int S,
                                    const float* __restrict__ GB,
                                    const float* __restrict__ sq,
                                    float* __restrict__ D) {
    const int TQ = PP / 16;                          // 64 q-tiles
    const int TQ4 = TQ / 4;                          // 16 strips
    int wave = (blockIdx.x * blockDim.x + threadIdx.x) >> 5;
    if (wave >= BB * TQ * TQ4) return;               // wave-uniform
    int b   = wave / (TQ * TQ4);
    int rem = wave % (TQ * TQ4);
    int pt  = rem / TQ4;
    int qt4 = rem % TQ4;
    int lane = threadIdx.x & 31, lane15 = lane & 15, hi = lane >> 4;
    int p0 = pt * 16, q0 = qt4 * 64;

    const float* arow = X + ((size_t)b * PP + p0 + lane15) * lda + 2 * hi;
    const v2f*   bp   = (const v2f*)GB + (((size_t)b * TQ + qt4 * 4) * S) * 32 + lane;
    const size_t TB = (size_t)S * 32;                // v2f per packed tile

    v8f acc0 = {}, acc1 = {}, acc2 = {}, acc3 = {};
    #pragma unroll 4
    for (int s = 0; s < S; ++s) {
        v2f a   = *(const v2f*)(arow + 4 * s);
        v2f bv0 = bp[(size_t)s * 32];
        v2f bv1 = bp[(size_t)s * 32 + TB];
        v2f bv2 = bp[(size_t)s * 32 + 2 * TB];
        v2f bv3 = bp[(size_t)s * 32 + 3 * TB];
        acc0 = __builtin_amdgcn_wmma_f32_16x16x4_f32(false, a, false, bv0, (short)0, acc0, false, false);
        acc1 = __builtin_amdgcn_wmma_f32_16x16x4_f32(false, a, false, bv1, (short)0, acc1, false, false);
        acc2 = __builtin_amdgcn_wmma_f32_16x16x4_f32(false, a, false, bv2, (short)0, acc2, false, false);
        acc3 = __builtin_amdgcn_wmma_f32_16x16x4_f32(false, a, false, bv3, (short)0, acc3, false, false);
    }
    const float* sqb = sq + (size_t)b * PP;
    float* Db = D + (size_t)b * PP * PP;
    v8f* accs[4] = { &acc0, &acc1, &acc2, &acc3 };
    for (int t = 0; t < 4; ++t) {
        float sqq = sqb[q0 + 16 * t + lane15];
        for (int r = 0; r < 8; ++r) {
            int p = p0 + r + 8 * hi;
            Db[(size_t)p * PP + q0 + 16 * t + lane15] = sqb[p] + sqq - 2.f * (*accs[t])[r];
        }
    }
}

// ---------------------------------------------------------------------------
// Per-point top-K smallest distances (self included).
// ---------------------------------------------------------------------------
__global__ void dg_topk_kernel(const float* __restrict__ D, int* __restrict__ idx) {
    int i = blockIdx.x * blockDim.x + threadIdx.x;
    if (i >= BB * PP) return;
    int b = i / PP, p = i % PP;
    const float* row = D + (size_t)b * PP * PP + (size_t)p * PP;
    float bd[KK];
    int   bi[KK];
    for (int k = 0; k < KK; ++k) { bd[k] = 3.4e38f; bi[k] = 0; }
    float worst = 3.4e38f;
    int   wpos  = 0;
    for (int q = 0; q < PP; ++q) {
        float d = row[q];
        if (d < worst) {
            bd[wpos] = d; bi[wpos] = q;
            worst = -3.4e38f;
            for (int k = 0; k < KK; ++k)
                if (bd[k] > worst) { worst = bd[k]; wpos = k; }
        }
    }
    int* oi = idx + (size_t)i * KK;
    for (int k = 0; k < KK; ++k) oi[k] = bi[k];
}

// ---------------------------------------------------------------------------
// fp32 WMMA GEMM with pre-packed B: C[M,N] = A[M,Kpad] * B (+bias).
// Each wave computes a 16x64 strip (4 N-tiles, 4 accumulators); one A load
// feeds 4 WMMAs per K-step. N must be a multiple of 64.
// ---------------------------------------------------------------------------
__global__ void dg_gemm_wmma_f32(const float* __restrict__ A, int lda,
                                 const float* __restrict__ Bp,
                                 const float* __restrict__ bias,
                                 float* __restrict__ Cm, int ldc,
                                 int M, int N, int Kpad) {
    int S      = Kpad >> 2;
    int nTn4   = N >> 6;                             // strips of 4 tiles
    int nTiles = (M >> 4) * nTn4;
    int wave   = (blockIdx.x * blockDim.x + threadIdx.x) >> 5;
    if (wave >= nTiles) return;                      // wave-uniform
    int mt  = wave / nTn4;
    int nt4 = wave % nTn4;
    int lane = threadIdx.x & 31, lane15 = lane & 15, hi = lane >> 4;
    int m0 = mt * 16, n0 = nt4 * 64;

    const float* arow = A + (size_t)(m0 + lane15) * lda + 2 * hi;
    const v2f*   bp   = (const v2f*)Bp + (size_t)(nt4 * 4) * S * 32 + lane;
    const size_t TB   = (size_t)S * 32;

    v8f acc0 = {}, acc1 = {}, acc2 = {}, acc3 = {};
    #pragma unroll 4
    for (int s = 0; s < S; ++s) {
        v2f a   = *(const v2f*)(arow + 4 * s);
        v2f bv0 = bp[(size_t)s * 32];
        v2f bv1 = bp[(size_t)s * 32 + TB];
        v2f bv2 = bp[(size_t)s * 32 + 2 * TB];
        v2f bv3 = bp[(size_t)s * 32 + 3 * TB];
        acc0 = __builtin_amdgcn_wmma_f32_16x16x4_f32(false, a, false, bv0, (short)0, acc0, false, false);
        acc1 = __builtin_amdgcn_wmma_f32_16x16x4_f32(false, a, false, bv1, (short)0, acc1, false, false);
        acc2 = __builtin_amdgcn_wmma_f32_16x16x4_f32(false, a, false, bv2, (short)0, acc2, false, false);
        acc3 = __builtin_amdgcn_wmma_f32_16x16x4_f32(false, a, false, bv3, (short)0, acc3, false, false);
    }
    v8f* accs[4] = { &acc0, &acc1, &acc2, &acc3 };
    for (int t = 0; t < 4; ++t) {
        int nc = n0 + 16 * t + lane15;
        float bvv = bias ? bias[nc] : 0.f;
        for (int r = 0; r < 8; ++r)
            Cm[(size_t)(m0 + r + 8 * hi) * ldc + nc] = (*accs[t])[r] + bvv;
    }
}

// ---------------------------------------------------------------------------
// EdgeConv epilogue: Out[p][c] = U[p][c] + bias[c] + max_k V[idx[p][k]][c]
// ---------------------------------------------------------------------------
__global__ void dg_edge_aggregate_kernel(const float* __restrict__ U,
                                         const float* __restrict__ V,
                                         const float* __restrict__ bias,
                                         const int* __restrict__ idx,
                                         float* __restrict__ Out,
                                         int Cout, int ldo) {
    int t = blockIdx.x * blockDim.x + threadIdx.x;
    if (t >= BB * PP * Cout) return;
    int c = t % Cout;
    int i = t / Cout;
    int b = i / PP;
    const int* ni = idx + (size_t)i * KK;
    float m = -3.4e38f;
    for (int k = 0; k < KK; ++k) {
        int q = ni[k];
        if (k + 1 < KK)
            __builtin_prefetch(&V[((size_t)b * PP + ni[k + 1]) * Cout + c], 0, 0);
        m = fmaxf(m, V[((size_t)b * PP + q) * Cout + c]);
    }
    Out[(size_t)i * ldo + c] = U[(size_t)i * Cout + c] + bias[c] + m;
}

// ---------------------------------------------------------------------------
// Global max+mean pool over P points per cloud.
// ---------------------------------------------------------------------------
__global__ void dg_pool_kernel(const float* __restrict__ Outm, float* __restrict__ H) {
    int t = blockIdx.x * blockDim.x + threadIdx.x;
    if (t >= BB * 1024) return;
    int b = t / 1024, c = t % 1024;
    const float* base = Outm + (size_t)b * PP * 1024 + c;
    float mx = -3.4e38f, sm = 0.f;
    for (int p = 0; p < PP; ++p) {
        float v = base[(size_t)p * 1024];
        mx = fmaxf(mx, v);
        sm += v;
    }
    H[(size_t)b * 2048 + c]        = mx;
    H[(size_t)b * 2048 + 1024 + c] = sm * (1.f / PP);
}

// ---------------------------------------------------------------------------
// Fused head layer: per-column GEMM over 8 rows + BatchNorm(axis=0) + LeakyReLU.
// ---------------------------------------------------------------------------
__global__ void dg_head_layer_kernel(const float* __restrict__ Hin, int Kdim,
                                     const float* __restrict__ W,
                                     const float* __restrict__ bias,
                                     const float* __restrict__ gamma,
                                     const float* __restrict__ beta,
                                     float* __restrict__ Hout, int N) {
    int c = blockIdx.x * blockDim.x + threadIdx.x;
    if (c >= N) return;
    float y[BB];
    for (int r = 0; r < BB; ++r) {
        float s = bias[c];
        const float* hr = Hin + (size_t)r * Kdim;
        for (int k = 0; k < Kdim; ++k) s += hr[k] * W[(size_t)k * N + c];
        y[r] = s;
    }
    float mu = 0.f;
    for (int r = 0; r < BB; ++r) mu += y[r];
    mu *= (1.f / BB);
    float var = 0.f;
    for (int r = 0; r < BB; ++r) { float d = y[r] - mu; var += d * d; }
    var *= (1.f / BB);
    float rs = rsqrtf(var + EPSF);
    float g = gamma[c], be = beta[c];
    for (int r = 0; r < BB; ++r) {
        float v = (y[r] - mu) * rs * g + be;
        Hout[(size_t)r * N + c] = (v >= 0.f) ? v : 0.2f * v;
    }
}

// ---------------------------------------------------------------------------
// Final projection: out[8,40] = H2[8,256] @ Wc + bc
// ---------------------------------------------------------------------------
__global__ void dg_head_out_kernel(const float* __restrict__ H2,
                                   const float* __restrict__ Wc,
                                   const float* __restrict__ bc,
                                   float* __restrict__ outp) {
    int t = blockIdx.x * blockDim.x + threadIdx.x;
    if (t >= BB * 40) return;
    int r = t / 40, c = t % 40;
    float s = bc[c];
    const float* hr = H2 + (size_t)r * 256;
    for (int k = 0; k < 256; ++k) s += hr[k] * Wc[(size_t)k * 40 + c];
    outp[t] = s;
}

// ---------------------------------------------------------------------------
extern "C" void kernel_launch(void* const* d_in, const int* in_sizes, int n_in,
                              void* d_out, int out_size, void* d_ws, size_t ws_size,
                              hipStream_t stream) {
    const float* pos = (const float*)d_in[0];
    const float* W1 = (const float*)d_in[2];  const float* b1 = (const float*)d_in[3];
    const float* W2 = (const float*)d_in[4];  const float* b2 = (const float*)d_in[5];
    const float* W3 = (const float*)d_in[6];  const float* b3 = (const float*)d_in[7];
    const float* W4 = (const float*)d_in[8];  const float* b4 = (const float*)d_in[9];
    const float* Wm = (const float*)d_in[10]; const float* bm = (const float*)d_in[11];
    const float* Wa = (const float*)d_in[12]; const float* ba = (const float*)d_in[13];
    const float* ga = (const float*)d_in[14]; const float* bea = (const float*)d_in[15];
    const float* Wb = (const float*)d_in[16]; const float* bb = (const float*)d_in[17];
    const float* gb = (const float*)d_in[18]; const float* beb = (const float*)d_in[19];
    const float* Wc = (const float*)d_in[20]; const float* bc = (const float*)d_in[21];

    // Workspace carve-up (floats).
    float* ws = (float*)d_ws;
    size_t o = 0;
    float* SQ  = ws + o; o += (size_t)BB * PP;                 // 8192
    int*   IDX = (int*)(ws + o); o += (size_t)BB * PP * KK;    // 163840
    float* XP  = ws + o; o += (size_t)BB * PP * 4;             // padded pos
    float* GB  = ws + o; o += (size_t)BB * PP * 128;           // gram B-pack (max)
    float* WDP = ws + o; o += (size_t)128 * 256;               // packed Wtop-Wbot
    float* WBP = ws + o; o += (size_t)128 * 256;               // packed Wbot
    float* WMP = ws + o; o += (size_t)512 * 1024;              // packed Wm
    float* U   = ws + o; o += (size_t)BB * PP * 256;
    float* V   = ws + o; o += (size_t)BB * PP * 256;
    float* XC  = ws + o; o += (size_t)BB * PP * 512;           // [x1|x2|x3|x4]
    float* D   = ws + o; o += (size_t)BB * PP * PP;            // aliased with OUTM
    float* OUTM = D;
    float* H   = ws + o; o += (size_t)BB * 2048;
    float* H1  = ws + o; o += (size_t)BB * 512;
    float* H2  = ws + o; o += (size_t)BB * 256;

    const int NPTS = BB * PP;                                  // 8192

    // cinPad: padded K (multiple of 4); cinValid: true weight row count.
    auto run_layer = [&](const float* X, int lda, int cinPad, int cinValid,
                         const float* W, const float* bias, int Cout,
                         float* Out, int ldo) {
        int S = cinPad >> 2;
        dg_sqnorm_kernel<<<NPTS / 256, 256, 0, stream>>>(X, lda, cinPad, SQ);
        int gpTot = BB * 64 * S * 64;
        dg_grampack_kernel<<<(gpTot + 255) / 256, 256, 0, stream>>>(X, lda, S, GB);
        // gram: 8 clouds * 64 p-tiles * 16 q-strips = 8192 waves -> 1024 blocks
        dg_gram_dist_kernel<<<1024, 256, 0, stream>>>(X, lda, S, GB, SQ, D);
        dg_topk_kernel<<<NPTS / 256, 256, 0, stream>>>(D, IDX);
        int bpTot = (Cout >> 4) * S * 64;
        dg_bpack_kernel<<<(bpTot + 255) / 256, 256, 0, stream>>>(
            W, W + (size_t)cinValid * Cout, Cout, cinValid, S, Cout, WDP);
        dg_bpack_kernel<<<(bpTot + 255) / 256, 256, 0, stream>>>(
            W + (size_t)cinValid * Cout, nullptr, Cout, cinValid, S, Cout, WBP);
        int tiles = (NPTS / 16) * (Cout / 64);                 // multiple of 8
        dg_gemm_wmma_f32<<<tiles / 8, 256, 0, stream>>>(
            X, lda, WDP, nullptr, U, Cout, NPTS, Cout, cinPad);
        dg_gemm_wmma_f32<<<tiles / 8, 256, 0, stream>>>(
            X, lda, WBP, nullptr, V, Cout, NPTS, Cout, cinPad);
        int tot = NPTS * Cout;
        dg_edge_aggregate_kernel<<<(tot + 255) / 256, 256, 0, stream>>>(
            U, V, bias, IDX, Out, Cout, ldo);
    };

    dg_pad_pos_kernel<<<NPTS / 256, 256, 0, stream>>>(pos, XP);

    run_layer(XP,       4,   4,   3,   W1, b1, 64,  XC + 0,   512);
    run_layer(XC + 0,   512, 64,  64,  W2, b2, 64,  XC + 64,  512);
    run_layer(XC + 64,  512, 64,  64,  W3, b3, 128, XC + 128, 512);
    run_layer(XC + 128, 512, 128, 128, W4, b4, 256, XC + 256, 512);

    // out = cat @ Wm + bm : [8192,512] x [512,1024]
    {
        int S = 512 >> 2;                                      // 128
        int bpTot = (1024 >> 4) * S * 64;                      // 524288
        dg_bpack_kernel<<<(bpTot + 255) / 256, 256, 0, stream>>>(
            Wm, nullptr, 1024, 512, S, 1024, WMP);
        int tiles = (NPTS / 16) * (1024 / 64);                 // 8192
        dg_gemm_wmma_f32<<<tiles / 8, 256, 0, stream>>>(
            XC, 512, WMP, bm, OUTM, 1024, NPTS, 1024, 512);
    }
    dg_pool_kernel<<<(BB * 1024 + 255) / 256, 256, 0, stream>>>(OUTM, H);
    dg_head_layer_kernel<<<2, 256, 0, stream>>>(H,  2048, Wa, ba, ga, bea, H1, 512);
    dg_head_layer_kernel<<<1, 256, 0, stream>>>(H1, 512,  Wb, bb, gb, beb, H2, 256);
    dg_head_out_kernel<<<2, 256, 0, stream>>>(H2, Wc, bc, (float*)d_out);
}